// SourceModule_18356690223311
// MI455X (gfx1250) — compile-verified
//
#include <hip/hip_runtime.h>
#include <math.h>
#include <stdint.h>

// ---------------- problem constants ----------------
#define H        8
#define FS       240              // frame shift
#define NFR      500              // frames
#define BB       32               // batch
#define TT       (NFR * FS)       // 120000 samples per row
#define ALPHA_F  0.1f
#define UV_SCALE 11.111111f       // ALPHA / (3*SIGMA)
#define W_SCALE_F ((float)(6.283185307179586 / 24000.0))  // 2*pi/SR
#define TWO_PI_D  6.283185307179586
#define INV_2PI_D 0.15915494309189535

// ---------------- tiling ----------------
#define TILE     128                          // samples per wave-tile (32 lanes x 4)
#define TPR      ((TT + TILE - 1) / TILE)     // 938 tiles per row (last partial)
#define NTILES   (BB * TPR)                   // 30016
#define THREADS  128
#define WPB      (THREADS / 32)               // 4 waves per block
#define BLOCKS   1024
#define NWAVES   (BLOCKS * WPB)               // 4096

// native clang vector types (accepted by the amdgcn builtins)
typedef int   v4i __attribute__((ext_vector_type(4)));
typedef float f4  __attribute__((ext_vector_type(4)));
typedef __attribute__((address_space(1))) v4i g_v4i;   // global
typedef __attribute__((address_space(3))) v4i l_v4i;   // LDS

// ---- CDNA5 async global -> LDS copy (ASYNCcnt-tracked) ----
__device__ __forceinline__ void async_copy_b128(const void* gsrc, unsigned lds_off) {
#if __has_builtin(__builtin_amdgcn_global_load_async_to_lds_b128)
  __builtin_amdgcn_global_load_async_to_lds_b128(
      (g_v4i*)(uintptr_t)gsrc, (l_v4i*)lds_off, 0, 0);
#else
  asm volatile("global_load_async_to_lds_b128 %0, %1, off"
               :: "v"(lds_off), "v"(gsrc)
               : "memory");
#endif
}

template <int N>
__device__ __forceinline__ void wait_asynccnt() {
#if __has_builtin(__builtin_amdgcn_s_wait_asynccnt)
  __builtin_amdgcn_s_wait_asynccnt(N);
#else
  asm volatile("s_wait_asynccnt %0" :: "i"(N) : "memory");
#endif
  asm volatile("" ::: "memory");   // keep LDS reads below the wait
}

// fast tanh: (e^2x - 1)/(e^2x + 1) on the hardware exp pipe; |x| clamped to 9
// (tanh(9) == 1.0f to within float ulp), avoids inf/inf.
__device__ __forceinline__ float fast_tanh(float x) {
  float cx = fminf(fmaxf(x, -9.0f), 9.0f);
  float e  = __expf(2.0f * cx);
  return (e - 1.0f) / (e + 1.0f);
}

// ---------------- Kernel A: per-row frame-level phase scan (parallel) ----------------
// S[b][F] = (sum over frames f<F of 240 * 2*pi*f0[b][f]/SR) mod 2*pi   (exclusive prefix)
// Hillis-Steele scan in LDS (doubles); mod 2*pi via trunc+fma (no iterative fmod).
__global__ void __launch_bounds__(512)
f0_scan_kernel(const float* __restrict__ f0s, float* __restrict__ S) {
  __shared__ double sh[512];
  const int b = blockIdx.x;
  const int f = threadIdx.x;

  double v = 0.0;
  if (f < NFR)
    v = (double)FS * (double)W_SCALE_F * (double)f0s[b * NFR + f];
  sh[f] = v;
  __syncthreads();

#pragma unroll
  for (int off = 1; off < 512; off <<= 1) {
    double t = (f >= off) ? sh[f - off] : 0.0;
    __syncthreads();
    sh[f] += t;
    __syncthreads();
  }

  if (f < NFR) {
    double excl = sh[f] - v;                    // exclusive prefix
    double k = trunc(excl * INV_2PI_D);         // v_trunc_f64 (exact: k <= ~2200)
    S[b * NFR + f] = (float)fma(-TWO_PI_D, k, excl);   // excl mod 2*pi
  }
}

// ---------------- Kernel B: streaming synth ----------------
__global__ void __launch_bounds__(THREADS)
synth_kernel(const float* __restrict__ f0s,
             const float* __restrict__ phi,
             const float* __restrict__ amplitude,
             const float* __restrict__ noise,
             const float* __restrict__ Sx,
             float* __restrict__ out) {
  __shared__ f4 smbuf[WPB][2][H][32];   // 32 KB, per-wave double buffer

  const int warp = threadIdx.x >> 5;
  const int lane = threadIdx.x & 31;
  const int wglobal = blockIdx.x * WPB + warp;

  const int tpw = (NTILES + NWAVES - 1) / NWAVES;   // 8
  const int first = wglobal * tpw;
  const int last  = (first + tpw < NTILES) ? (first + tpw) : NTILES;
  if (first >= last) return;   // uniform per wave

  // uniform scalars (compiler scalarizes these)
  float amp[H + 1], ph[H];
#pragma unroll
  for (int i = 0; i <= H; ++i) amp[i] = amplitude[i];
#pragma unroll
  for (int i = 0; i < H; ++i)  ph[i]  = phi[i];

  auto issue_tile = [&](int tile, int buf) {
    int b  = tile / TPR;
    int rt = tile - b * TPR;
    int t0 = rt * TILE + lane * 4;
    if (t0 < TT) {
      const float* g = noise + ((size_t)b * TT + t0);
      unsigned l0 = (unsigned)(uintptr_t)&smbuf[warp][buf][0][lane];
#pragma unroll
      for (int h = 0; h < H; ++h)
        async_copy_b128(g + (size_t)h * ((size_t)BB * TT),
                        l0 + (unsigned)(h * 32 * (int)sizeof(f4)));
    }
  };

  auto compute_tile = [&](int tile, int buf) {
    int b  = tile / TPR;
    int rt = tile - b * TPR;
    int t0 = rt * TILE + lane * 4;
    if (t0 >= TT) return;
    int F = t0 / FS;
    int o = t0 - F * FS;                       // quad never straddles a frame (4 | 240)
    float f0 = f0s[b * NFR + F];
    float w  = W_SCALE_F * f0;
    float p0 = Sx[b * NFR + F] + (float)(o + 1) * w;   // phase of first sample, mod-2pi base
    bool voiced = (f0 > 0.0f);
    float sscale = voiced ? ALPHA_F : 0.0f;
    float nscale = voiced ? 1.0f : UV_SCALE;

    float acc0 = amp[H], acc1 = amp[H], acc2 = amp[H], acc3 = amp[H];
#pragma unroll
    for (int h = 0; h < H; ++h) {
      f4 n4 = smbuf[warp][buf][h][lane];   // exactly the bytes this lane async-loaded
      float k    = (float)(h + 1);
      float a    = amp[h];
      float base = k * p0 + ph[h];
      float kw   = k * w;
      acc0 += a * (sscale * __sinf(base)             + nscale * n4.x);
      acc1 += a * (sscale * __sinf(base + kw)        + nscale * n4.y);
      acc2 += a * (sscale * __sinf(base + 2.0f * kw) + nscale * n4.z);
      acc3 += a * (sscale * __sinf(base + 3.0f * kw) + nscale * n4.w);
    }
    f4 r;
    r.x = fast_tanh(acc0); r.y = fast_tanh(acc1);
    r.z = fast_tanh(acc2); r.w = fast_tanh(acc3);
    __builtin_nontemporal_store(r, (f4*)(out + (size_t)b * TT + t0));
  };

  // double-buffered pipeline: issue next tile's 8 async b128 loads, wait for the
  // previous 8 (ASYNCcnt decrements in order), compute from LDS.
  int buf = 0;
  issue_tile(first, 0);
  for (int t = first; t < last; ++t) {
    if (t + 1 < last) {
      issue_tile(t + 1, buf ^ 1);
      wait_asynccnt<8>();      // first 8 (current tile) complete
    } else {
      wait_asynccnt<0>();
    }
    compute_tile(t, buf);
    buf ^= 1;
  }
}

// ---------------- host entry ----------------
extern "C" void kernel_launch(void* const* d_in, const int* in_sizes, int n_in,
                              void* d_out, int out_size, void* d_ws, size_t ws_size,
                              hipStream_t stream) {
  const float* f0s       = (const float*)d_in[0];   // [32, 500]
  const float* phi       = (const float*)d_in[1];   // [8]
  const float* amplitude = (const float*)d_in[2];   // [9]
  const float* noise     = (const float*)d_in[3];   // [8, 32, 120000]
  float* out = (float*)d_out;                       // [32, 120000]
  float* S   = (float*)d_ws;                        // 32*500 floats = 64 KB scratch

  hipLaunchKernelGGL(f0_scan_kernel, dim3(BB), dim3(512), 0, stream, f0s, S);
  hipLaunchKernelGGL(synth_kernel, dim3(BLOCKS), dim3(THREADS), 0, stream,
                     f0s, phi, amplitude, noise, S, out);
}